// TransductiveGNN_74612171866465
// MI455X (gfx1250) — compile-verified
//
#include <hip/hip_runtime.h>
#include <hip/hip_bf16.h>

#define N_NODES  100000
#define N_EDGES  3200000
#define N_FEAT   128
#define EMB      64
#define HID      64
#define N_SUB    500000
#define N_GRAPHS 1024
#define BN_EPS   1e-5f

typedef __attribute__((ext_vector_type(2))) float v2f;
typedef __attribute__((ext_vector_type(8))) float v8f;

// ---------------- init: zero accumulators, deg=1 (self-loop) ----------------
__global__ void k_init(float* __restrict__ deg, float* __restrict__ agg,
                       float* __restrict__ sums, float* __restrict__ cnts) {
  int i = blockIdx.x * blockDim.x + threadIdx.x;
  if (i < N_NODES * EMB) agg[i] = 0.0f;
  if (i < N_NODES)       deg[i] = 1.0f;          // self-loop contributes 1
  if (i < N_GRAPHS * EMB) sums[i] = 0.0f;
  if (i < N_GRAPHS)       cnts[i] = 0.0f;
}

// ---------------- degree accumulation over edge targets ----------------
__global__ void k_degree(const int* __restrict__ col, float* __restrict__ deg) {
  int e = blockIdx.x * blockDim.x + threadIdx.x;
  if (e < N_EDGES) atomicAdd(&deg[col[e]], 1.0f);
}

__global__ void k_rsqrt(float* __restrict__ deg) {
  int i = blockIdx.x * blockDim.x + threadIdx.x;
  if (i < N_NODES) deg[i] = rsqrtf(deg[i]);      // deg >= 1 always
}

// ---------------- WMMA GEMM: XW[100000x64] = X[100000x128] @ W[128x64] ------
// one wave per 16-row tile; 4 N-tiles of 16; K loop in steps of 4 (f32 WMMA)
__global__ void k_gemm_xw(const float* __restrict__ X, const float* __restrict__ W,
                          float* __restrict__ XW) {
  int wave = (blockIdx.x * blockDim.x + threadIdx.x) >> 5;
  int lane = threadIdx.x & 31;
  if (wave >= N_NODES / 16) return;              // whole-wave exit: EXEC stays all-1s
  const int row0 = wave * 16;
  const int half = lane >> 4;                    // K-half selector
  const int l    = lane & 15;                    // M (for A) / N (for B,C)
  v8f c0 = {}, c1 = {}, c2 = {}, c3 = {};
  const float* xrow = X + (long long)(row0 + l) * N_FEAT;
  for (int k = 0; k < N_FEAT; k += 4) {
    int ka = k + half * 2;
    v2f a; a.x = xrow[ka]; a.y = xrow[ka + 1];
    const float* w0 = W + (long long)ka * EMB + l;
    const float* w1 = W + (long long)(ka + 1) * EMB + l;
    v2f b0, b1, b2, b3;
    b0.x = w0[0];  b0.y = w1[0];
    b1.x = w0[16]; b1.y = w1[16];
    b2.x = w0[32]; b2.y = w1[32];
    b3.x = w0[48]; b3.y = w1[48];
    c0 = __builtin_amdgcn_wmma_f32_16x16x4_f32(false, a, false, b0, (short)0, c0, false, false);
    c1 = __builtin_amdgcn_wmma_f32_16x16x4_f32(false, a, false, b1, (short)0, c1, false, false);
    c2 = __builtin_amdgcn_wmma_f32_16x16x4_f32(false, a, false, b2, (short)0, c2, false, false);
    c3 = __builtin_amdgcn_wmma_f32_16x16x4_f32(false, a, false, b3, (short)0, c3, false, false);
  }
  for (int r = 0; r < 8; ++r) {
    int m = row0 + half * 8 + r;                 // C layout: upper lane half -> M+8
    float* o = XW + (long long)m * EMB + l;
    o[0] = c0[r]; o[16] = c1[r]; o[32] = c2[r]; o[48] = c3[r];
  }
}

// ---------------- edge scatter: agg[col] += xw[row] * dinv[row]*dinv[col] ---
__global__ void k_scatter(const int* __restrict__ rows, const int* __restrict__ cols,
                          const float* __restrict__ dinv, const float* __restrict__ xw,
                          float* __restrict__ agg) {
  int tid = blockIdx.x * blockDim.x + threadIdx.x;   // N_EDGES*16 threads
  int e = tid >> 4, q = tid & 15;
  if (e >= N_EDGES) return;
  int r = rows[e], c = cols[e];
  float nrm = dinv[r] * dinv[c];
  const float4 v = *(const float4*)(xw + (long long)r * EMB + q * 4);
  float* o = agg + (long long)c * EMB + q * 4;
  atomicAdd(o + 0, v.x * nrm);
  atomicAdd(o + 1, v.y * nrm);
  atomicAdd(o + 2, v.z * nrm);
  atomicAdd(o + 3, v.w * nrm);
}

// ---------------- finalize: add self-loop term + bias, ReLU (in place) ------
__global__ void k_finalize(const float* __restrict__ xw, const float* __restrict__ dinv,
                           const float* __restrict__ b, float* __restrict__ agg) {
  int i = blockIdx.x * blockDim.x + threadIdx.x;
  if (i >= N_NODES * EMB) return;
  int n = i >> 6, f = i & 63;
  float d = dinv[n];
  float v = agg[i] + xw[i] * d * d + b[f];
  agg[i] = fmaxf(v, 0.0f);
}

// ---------------- gather + segment-sum pool ----------------
__global__ void k_pool_feat(const int* __restrict__ subidx, const int* __restrict__ batch,
                            const float* __restrict__ h, float* __restrict__ sums) {
  int tid = blockIdx.x * blockDim.x + threadIdx.x;   // N_SUB*16 threads
  int i = tid >> 4, q = tid & 15;
  if (i >= N_SUB) return;
  int node = subidx[i], g = batch[i];
  const float4 v = *(const float4*)(h + (long long)node * EMB + q * 4);
  float* o = sums + g * EMB + q * 4;
  atomicAdd(o + 0, v.x);
  atomicAdd(o + 1, v.y);
  atomicAdd(o + 2, v.z);
  atomicAdd(o + 3, v.w);
}

__global__ void k_pool_cnt(const int* __restrict__ batch, float* __restrict__ cnts) {
  int i = blockIdx.x * blockDim.x + threadIdx.x;
  if (i < N_SUB) atomicAdd(&cnts[batch[i]], 1.0f);
}

__global__ void k_pool_div(const float* __restrict__ sums, const float* __restrict__ cnts,
                           float* __restrict__ pooled) {
  int i = blockIdx.x * blockDim.x + threadIdx.x;
  if (i >= N_GRAPHS * EMB) return;
  pooled[i] = sums[i] / fmaxf(cnts[i >> 6], 1.0f);
}

// ---------------- WMMA GEMM: Z = relu(P[1024x64] @ W1[64x64] + b1) ----------
__global__ void k_gemm_mlp(const float* __restrict__ P, const float* __restrict__ W1,
                           const float* __restrict__ b1, float* __restrict__ Z) {
  int wave = (blockIdx.x * blockDim.x + threadIdx.x) >> 5;
  int lane = threadIdx.x & 31;
  if (wave >= N_GRAPHS / 16) return;
  const int row0 = wave * 16;
  const int half = lane >> 4;
  const int l    = lane & 15;
  v8f c0 = {}, c1 = {}, c2 = {}, c3 = {};
  const float* prow = P + (long long)(row0 + l) * EMB;
  for (int k = 0; k < EMB; k += 4) {
    int ka = k + half * 2;
    v2f a; a.x = prow[ka]; a.y = prow[ka + 1];
    const float* w0 = W1 + (long long)ka * HID + l;
    const float* w1 = W1 + (long long)(ka + 1) * HID + l;
    v2f b0, b1v, b2, b3;
    b0.x  = w0[0];  b0.y  = w1[0];
    b1v.x = w0[16]; b1v.y = w1[16];
    b2.x  = w0[32]; b2.y  = w1[32];
    b3.x  = w0[48]; b3.y  = w1[48];
    c0 = __builtin_amdgcn_wmma_f32_16x16x4_f32(false, a, false, b0,  (short)0, c0, false, false);
    c1 = __builtin_amdgcn_wmma_f32_16x16x4_f32(false, a, false, b1v, (short)0, c1, false, false);
    c2 = __builtin_amdgcn_wmma_f32_16x16x4_f32(false, a, false, b2,  (short)0, c2, false, false);
    c3 = __builtin_amdgcn_wmma_f32_16x16x4_f32(false, a, false, b3,  (short)0, c3, false, false);
  }
  float bb0 = b1[l], bb1 = b1[l + 16], bb2 = b1[l + 32], bb3 = b1[l + 48];
  for (int r = 0; r < 8; ++r) {
    int m = row0 + half * 8 + r;
    float* o = Z + (long long)m * HID + l;
    o[0]  = fmaxf(c0[r] + bb0, 0.0f);
    o[16] = fmaxf(c1[r] + bb1, 0.0f);
    o[32] = fmaxf(c2[r] + bb2, 0.0f);
    o[48] = fmaxf(c3[r] + bb3, 0.0f);
  }
}

// ---------------- batch-norm stats (training mode, biased var) --------------
__global__ void k_bnstats(const float* __restrict__ Z, float* __restrict__ mu,
                          float* __restrict__ var) {
  __shared__ float ssum[256], ssq[256];
  int h = blockIdx.x;                  // one block per feature column (64 blocks)
  int t = threadIdx.x;
  float s = 0.0f, q = 0.0f;
  for (int g = t; g < N_GRAPHS; g += 256) {
    float v = Z[g * HID + h];
    s += v; q += v * v;
  }
  ssum[t] = s; ssq[t] = q;
  __syncthreads();
  for (int o = 128; o > 0; o >>= 1) {
    if (t < o) { ssum[t] += ssum[t + o]; ssq[t] += ssq[t + o]; }
    __syncthreads();
  }
  if (t == 0) {
    float m = ssum[0] * (1.0f / N_GRAPHS);
    mu[h]  = m;
    var[h] = ssq[0] * (1.0f / N_GRAPHS) - m * m;
  }
}

// ---------------- normalize + final linear [HID -> 1] -----------------------
__global__ void k_final(const float* __restrict__ Z, const float* __restrict__ mu,
                        const float* __restrict__ var, const float* __restrict__ gamma,
                        const float* __restrict__ beta, const float* __restrict__ W2,
                        const float* __restrict__ b2, float* __restrict__ out) {
  int g = blockIdx.x * blockDim.x + threadIdx.x;
  if (g >= N_GRAPHS) return;
  float acc = 0.0f;
  for (int h = 0; h < HID; ++h) {
    float zn = (Z[g * HID + h] - mu[h]) * rsqrtf(var[h] + BN_EPS) * gamma[h] + beta[h];
    acc += zn * W2[h];
  }
  out[g] = acc + b2[0];
}

extern "C" void kernel_launch(void* const* d_in, const int* in_sizes, int n_in,
                              void* d_out, int out_size, void* d_ws, size_t ws_size,
                              hipStream_t stream) {
  const float* full_x = (const float*)d_in[0];
  const int*   ei     = (const int*)d_in[1];   // [2, N_EDGES]: rows then cols
  const int*   subidx = (const int*)d_in[2];
  const int*   batch  = (const int*)d_in[3];
  const float* W_enc  = (const float*)d_in[4];
  const float* b_enc  = (const float*)d_in[5];
  const float* W1     = (const float*)d_in[6];
  const float* b1     = (const float*)d_in[7];
  const float* gamma  = (const float*)d_in[8];
  const float* beta   = (const float*)d_in[9];
  const float* W2     = (const float*)d_in[10];
  const float* b2     = (const float*)d_in[11];
  float* out = (float*)d_out;

  const int* e_row = ei;
  const int* e_col = ei + N_EDGES;

  // workspace layout (floats, 256-aligned offsets)
  float* ws = (float*)d_ws;
  float* dinv   = ws;                              // 100000 (deg -> dinv in place)
  float* xw     = ws + 100352;                     // 6.4M
  float* agg    = xw + (size_t)N_NODES * EMB;      // 6.4M (becomes h after finalize)
  float* sums   = agg + (size_t)N_NODES * EMB;     // 65536
  float* cnts   = sums + N_GRAPHS * EMB;           // 1024
  float* pooled = cnts + 1024;                     // 65536
  float* z      = pooled + N_GRAPHS * EMB;         // 65536
  float* mu     = z + N_GRAPHS * HID;              // 64
  float* var    = mu + 64;                         // 64

  const int B = 256;

  // 1) zero accumulators / init degree with self-loop
  k_init<<<(N_NODES * EMB + B - 1) / B, B, 0, stream>>>(dinv, agg, sums, cnts);

  // 2) degree -> dinv
  k_degree<<<(N_EDGES + B - 1) / B, B, 0, stream>>>(e_col, dinv);
  k_rsqrt<<<(N_NODES + B - 1) / B, B, 0, stream>>>(dinv);

  // 3) xw = x @ W_enc  (WMMA f32 16x16x4; 6250 M-tiles, 4 waves/block)
  k_gemm_xw<<<(N_NODES / 16 + 3) / 4, 128, 0, stream>>>(full_x, W_enc, xw);

  // 4) edge message scatter (dominant memory-bound phase)
  {
    long long t = (long long)N_EDGES * 16;
    k_scatter<<<(int)((t + B - 1) / B), B, 0, stream>>>(e_row, e_col, dinv, xw, agg);
  }

  // 5) self-loop + bias + ReLU -> h (in agg)
  k_finalize<<<(N_NODES * EMB + B - 1) / B, B, 0, stream>>>(xw, dinv, b_enc, agg);

  // 6) gather + segment-mean pool
  {
    long long t = (long long)N_SUB * 16;
    k_pool_feat<<<(int)((t + B - 1) / B), B, 0, stream>>>(subidx, batch, agg, sums);
  }
  k_pool_cnt<<<(N_SUB + B - 1) / B, B, 0, stream>>>(batch, cnts);
  k_pool_div<<<(N_GRAPHS * EMB + B - 1) / B, B, 0, stream>>>(sums, cnts, pooled);

  // 7) z = relu(pooled @ W1 + b1)  (WMMA, 64 M-tiles -> 16 blocks of 4 waves)
  k_gemm_mlp<<<N_GRAPHS / 16 / 4, 128, 0, stream>>>(pooled, W1, b1, z);

  // 8) batch-norm stats + final projection
  k_bnstats<<<HID, 256, 0, stream>>>(z, mu, var);
  k_final<<<(N_GRAPHS + B - 1) / B, B, 0, stream>>>(z, mu, var, gamma, beta, W2, b2, out);
}